// MarkovChainEvolutionModel_74053826117693
// MI455X (gfx1250) — compile-verified
//
#include <hip/hip_runtime.h>
#include <hip/hip_bf16.h>
#include <math.h>

// Problem constants (from reference)
#define B_DIM 64
#define L_DIM 16384
#define ORDER 3
#define V_DIM 4
#define CTX_DIM 64   // V^ORDER
#define NBITS 16

typedef float v2f __attribute__((ext_vector_type(2)));
typedef float v8f __attribute__((ext_vector_type(8)));

// ---------------------------------------------------------------------------
// Kernel 1: compute table[chain=b*64+ctx][cur][c] = (T[ctx]^n(b,ctx))[cur][c]
// One wave handles 4 chains via block-diagonal packing into 16x16x4 f32 WMMA.
// 4096 chains total -> 1024 waves -> 256 blocks of 128 threads (4 waves).
// ---------------------------------------------------------------------------
__global__ __launch_bounds__(128) void mc_pow_kernel(
    const float* __restrict__ time, const float* __restrict__ T,
    const float* __restrict__ W1,   const float* __restrict__ b1,
    const float* __restrict__ W2,   const float* __restrict__ b2,
    float* __restrict__ tbl)
{
    const int lane = threadIdx.x & 31;
    const int wave = threadIdx.x >> 5;
    const int globalWave = blockIdx.x * 4 + wave;
    const int chainBase  = globalWave * 4;          // 4 chains per wave
    const int b = chainBase >> 6;                   // same b for all 4 chains
    const float t = time[b];

    // Per-lane redundant scalar compute of n for the wave's 4 chains
    // (rate = softplus(relu(t*W1+b1) @ W2[ctx] + b2[ctx]); n = clip(int(rate*t+1)))
    int nexp[4];
#pragma unroll
    for (int q = 0; q < 4; ++q) {
        const int ctx = (chainBase + q) & 63;
        float acc = b2[ctx];
        for (int j = 0; j < 32; ++j) {
            float h = fmaf(t, W1[j], b1[j]);
            h = fmaxf(h, 0.0f);
            acc = fmaf(h, W2[ctx * 32 + j], acc);
        }
        // softplus = log1p(exp(-|x|)) + max(x,0)
        float sp = log1pf(expf(-fabsf(acc))) + fmaxf(acc, 0.0f);
        float nf = fmaf(sp, t, 1.0f);
        int ni = (int)nf;                 // trunc toward zero, nf > 0
        ni = ni < 0 ? 0 : (ni > 65535 ? 65535 : ni);
        nexp[q] = ni;
    }

    // WMMA lane mapping (wave32):
    //  A 16x4 f32 : vgpr v, lane l -> (M = l&15, K = 2*(l>>4) + v)
    //  B 4x16 f32 : vgpr v, lane l -> (K = 2*(l>>4) + v, N = l&15)
    //  D 16x16 f32: vgpr r, lane l -> (M = r + 8*(l>>4), N = l&15)
    const int m  = lane & 15;     // A row == B/D column for this lane
    const int hi = lane >> 4;
    const int k0 = hi * 2;        // K base {0,2}
    const int qA = m >> 2;        // which of the 4 chains this row/col belongs to
    const int rA = m & 3;         // local row/col within the 4x4 block

    // Initial M = T[ctx_q]; R = I (block-diagonal identity)
    const float* Mq = T + (((chainBase + qA) & 63) * 16);
    v2f aM = { Mq[rA * 4 + k0], Mq[rA * 4 + k0 + 1] };        // A = stacked M's
    v2f bM = { Mq[k0 * 4 + rA], Mq[(k0 + 1) * 4 + rA] };      // B = side-by-side M's
    v2f aR = { (rA == k0) ? 1.0f : 0.0f, (rA == k0 + 1) ? 1.0f : 0.0f };

    __shared__ float lds[4][2][256];   // [wave][{MM,RM}][16x16]
    float* mm = &lds[wave][0][0];
    float* rm = &lds[wave][1][0];

    for (int k = 0; k < NBITS; ++k) {
        v8f zero = {};
        // Diagonal blocks of dRM = R_q @ M_q ; dMM = M_q @ M_q
        v8f dRM = __builtin_amdgcn_wmma_f32_16x16x4_f32(
            false, aR, false, bM, (short)0, zero, false, false);
        v8f dMM = __builtin_amdgcn_wmma_f32_16x16x4_f32(
            false, aM, false, bM, (short)0, zero, false, false);

        // Stage D tiles through LDS to rebuild A/B layouts (diag blocks only)
#pragma unroll
        for (int r = 0; r < 8; ++r) {
            const int mr = r + 8 * hi;
            mm[mr * 16 + m] = dMM[r];
            rm[mr * 16 + m] = dRM[r];
        }
        __syncthreads();

        const int base = 4 * qA + k0;
        aM[0] = mm[m * 16 + base];
        aM[1] = mm[m * 16 + base + 1];
        bM[0] = mm[base * 16 + m];
        bM[1] = mm[(base + 1) * 16 + m];
        if ((nexp[qA] >> k) & 1) {      // conditional accumulate R <- R@M
            aR[0] = rm[m * 16 + base];
            aR[1] = rm[m * 16 + base + 1];
        }
        __syncthreads();
    }

    // Write final R (held in A layout): table[chain][row][col], row == "cur"
    const int chain = chainBase + qA;
    tbl[chain * 16 + rA * 4 + k0]     = aR[0];
    tbl[chain * 16 + rA * 4 + k0 + 1] = aR[1];
}

// ---------------------------------------------------------------------------
// Kernel 2: bandwidth-bound gather.
// out[b, p, :] = (p < 3) ? 0 : table[b][16*s[p-3]+4*s[p-2]+s[p-1]][s[p-1]][:]
// Block = 256 positions of one b; per-b table (4KB) + seq window in LDS.
// ---------------------------------------------------------------------------
__global__ __launch_bounds__(256) void mc_gather_kernel(
    const int* __restrict__ seq, const float* __restrict__ tbl,
    float* __restrict__ out)
{
    const int tid = threadIdx.x;
    const int b   = blockIdx.y;
    const int p0  = blockIdx.x * 256;

    __shared__ float s_tbl[CTX_DIM * V_DIM * V_DIM];   // 1024 floats = 4KB
    __shared__ int   s_seq[256 + ORDER];

    // Load this b's table slice (coalesced b128 loads)
    ((float4*)s_tbl)[tid] = ((const float4*)(tbl + b * 1024))[tid];

    // Load sequence window [p0-3, p0+255]
    for (int i = tid; i < 256 + ORDER; i += 256) {
        const int p = p0 - ORDER + i;
        s_seq[i] = (p >= 0) ? seq[b * L_DIM + p] : 0;
    }
    __syncthreads();

    const int p = p0 + tid;
    float4 res = make_float4(0.0f, 0.0f, 0.0f, 0.0f);
    if (p >= ORDER) {
        const int s0 = s_seq[tid];         // seq[p-3]
        const int s1 = s_seq[tid + 1];     // seq[p-2]
        const int s2 = s_seq[tid + 2];     // seq[p-1]
        const int ctx = s0 * 16 + s1 * 4 + s2;
        res = ((const float4*)s_tbl)[ctx * 4 + s2];
    }
    ((float4*)out)[b * L_DIM + p] = res;   // global_store_b128
}

extern "C" void kernel_launch(void* const* d_in, const int* in_sizes, int n_in,
                              void* d_out, int out_size, void* d_ws, size_t ws_size,
                              hipStream_t stream) {
    const int*   seq  = (const int*)d_in[0];
    const float* time = (const float*)d_in[1];
    const float* T    = (const float*)d_in[2];
    const float* W1   = (const float*)d_in[3];
    const float* b1   = (const float*)d_in[4];
    const float* W2   = (const float*)d_in[5];
    const float* b2   = (const float*)d_in[6];
    float* out = (float*)d_out;
    float* tbl = (float*)d_ws;   // 4096 chains * 16 floats = 256 KB

    // Phase 1: 4096 matrix-power chains, 4 per wave via WMMA (256 blocks x 128 thr)
    hipLaunchKernelGGL(mc_pow_kernel, dim3(4096 / 16), dim3(128), 0, stream,
                       time, T, W1, b1, W2, b2, tbl);

    // Phase 2: gather, 64x64 blocks of 256 positions
    hipLaunchKernelGGL(mc_gather_kernel, dim3(L_DIM / 256, B_DIM), dim3(256), 0, stream,
                       seq, tbl, out);
}